// MaskConsistencyLoss_83236466196484
// MI455X (gfx1250) — compile-verified
//
#include <hip/hip_runtime.h>
#include <hip/hip_bf16.h>

// Problem constants (from reference): x (8,256,128,128) f32, mask (8,1,128,128) f32
#define BB   8
#define CC   256
#define HW   16384
#define NSEG 256
#define BG   255
// 1 / (B*C*HW) = 1 / 33554432
#define INV_N 2.98023223876953125e-8f

typedef __attribute__((ext_vector_type(4))) unsigned int v4u_t;
typedef __attribute__((ext_vector_type(8))) int          v8i_t;
typedef __attribute__((ext_vector_type(4))) int          v4i_t;

// ---------------------------------------------------------------------------
// 1-D TDM load: DMA `nelem` f32 from global `gaddr` into LDS byte offset
// `lds_off`. Builds the Tensor DMA Descriptor (D#) per CDNA5 ISA ch.8:
//   group0: count=1 | lds_addr | global_addr[56:0] | type=2
//   group1: data_size=4B, tensor_dim0 = tile_dim0 = nelem (exact, no OOB)
//   group2/3: zero (1-D tensor)
// ---------------------------------------------------------------------------
__device__ __attribute__((always_inline)) inline
void tdm_load_row_f32(unsigned lds_off, const float* gaddr, unsigned nelem) {
  unsigned long long ga = (unsigned long long)(const void*)gaddr;
  v4u_t g0;
  g0.x = 1u;                                        // count=1, user mode, no gather
  g0.y = lds_off;                                   // lds_addr (bytes)
  g0.z = (unsigned)(ga & 0xFFFFFFFFu);              // global_addr[31:0]
  g0.w = (unsigned)((ga >> 32) & 0x01FFFFFFu)       // global_addr[56:32]
       | 0x80000000u;                               // type = 2 ("image")
  v8i_t g1;
  g1[0] = (int)(2u << 16);                          // wg_mask=0, data_size=2 (4B)
  g1[1] = (int)((nelem & 0xFFFFu) << 16);           // tensor_dim0[15:0] @ bits 63:48
  g1[2] = (int)((nelem >> 16) | (1u << 16));        // tensor_dim0[31:16], tensor_dim1=1
  g1[3] = (int)((nelem & 0xFFFFu) << 16);           // tile_dim0 @ bits 127:112
  g1[4] = 0;                                        // tile_dim1=0, tile_dim2=0 (1-D)
  g1[5] = (int)nelem;                               // tensor_dim0_stride[31:0]
  g1[6] = 0;
  g1[7] = 0;
  v4i_t g2 = {0, 0, 0, 0};
  v4i_t g3 = {0, 0, 0, 0};
#if __clang_major__ >= 23
  v8i_t g4 = {0, 0, 0, 0, 0, 0, 0, 0};
  __builtin_amdgcn_tensor_load_to_lds(g0, g1, g2, g3, g4, 0);
#else
  __builtin_amdgcn_tensor_load_to_lds(g0, g1, g2, g3, 0);
#endif
}

// ---------------------------------------------------------------------------
// Kernel 1: zero counts table + output accumulator (runs every launch)
// ---------------------------------------------------------------------------
__global__ void mcl_init_kernel(float* __restrict__ counts, float* __restrict__ out) {
  int i = blockIdx.x * 256 + threadIdx.x;
  if (i < BB * NSEG) counts[i] = 0.0f;
  if (i == 0) out[0] = 0.0f;
}

// ---------------------------------------------------------------------------
// Kernel 2: per-(batch, seg) pixel counts from mask (tiny: 131072 ids).
// LDS histogram per block, then one float atomic per (block, seg).
// Grid: 64 blocks (8 per batch), 256 threads.
// ---------------------------------------------------------------------------
__global__ __launch_bounds__(256) void mcl_count_kernel(
    const float* __restrict__ mask, float* __restrict__ counts) {
  __shared__ unsigned hist[NSEG];
  const int tid = threadIdx.x;
  hist[tid] = 0u;
  __syncthreads();
  const int bi    = blockIdx.x >> 3;
  const int chunk = blockIdx.x & 7;
  const float* mrow = mask + (size_t)bi * HW + (size_t)chunk * 2048;
#pragma unroll
  for (int k = 0; k < 8; ++k) {
    int id = (int)mrow[tid + (k << 8)];
    atomicAdd(&hist[id], 1u);
  }
  __syncthreads();
  atomicAdd(&counts[bi * NSEG + tid], (float)hist[tid]);
}

// ---------------------------------------------------------------------------
// Kernel 3: main single-pass kernel. One block per (batch, channel) row.
//  - 8 waves each TDM-load their 8KB slice of the x row into LDS
//  - sigmoid + LDS ds_add_f32 segment sums + local sum of sig^2 (non-bg)
//  - block contribution = sum(sig^2) - sum_s segsum[s]^2 / count[s]
//  - tree-reduce, one global f32 atomic of contribution * (1/N) into out
// ---------------------------------------------------------------------------
__global__ __launch_bounds__(256) void mcl_main_kernel(
    const float* __restrict__ x, const float* __restrict__ mask,
    const float* __restrict__ counts, float* __restrict__ out) {
  __shared__ float xbuf[HW];      // 64 KB: full (b,ch) row of x
  __shared__ float segsum[NSEG];  // per-segment sigmoid sums
  __shared__ float red[256];      // reduction scratch

  const int tid = threadIdx.x;
  const int blk = blockIdx.x;     // 0 .. B*C-1
  const int bi  = blk >> 8;       // / C
  const int ch  = blk & 255;      // % C

  segsum[tid] = 0.0f;

  // --- TDM stage: wave w DMAs elements [w*2048, (w+1)*2048) of this row ---
  unsigned w = (unsigned)__builtin_amdgcn_readfirstlane((int)((unsigned)tid >> 5));
  const float* src = x + (((size_t)(bi * CC + ch)) << 14) + (size_t)w * 2048u;
  unsigned lds_off = (unsigned)(size_t)(&xbuf[0]) + w * 2048u * 4u;
  tdm_load_row_f32(lds_off, src, 2048u);
  __builtin_amdgcn_s_wait_tensorcnt(0);
  asm volatile("" ::: "memory");
  __syncthreads();   // xbuf + segsum visible to all waves

  // --- accumulate stage ---
  const float* mrow = mask + (size_t)bi * HW;
  float sq = 0.0f;
#pragma unroll 4
  for (int k = 0; k < 64; ++k) {
    int p    = tid + (k << 8);
    float xv = xbuf[p];
    int id   = (int)mrow[p];
    float s  = 1.0f / (1.0f + __expf(-xv));   // sigmoid
    if (id != BG) {                            // bg: sig and label both zeroed -> 0
      atomicAdd(&segsum[id], s);               // ds_add_f32
      sq += s * s;
    }
  }
  __syncthreads();

  // --- finalize: v = sum(sig^2) - sum_{s<255} segsum^2 / count ---
  float v = sq;
  if (tid < BG) {
    float c  = counts[bi * NSEG + tid];
    float ss = segsum[tid];
    if (c > 0.0f) v -= ss * ss / c;
  }
  red[tid] = v;
  __syncthreads();
#pragma unroll
  for (int off = 128; off > 0; off >>= 1) {
    if (tid < off) red[tid] += red[tid + off];
    __syncthreads();
  }
  if (tid == 0) atomicAdd(out, red[0] * INV_N);
}

// ---------------------------------------------------------------------------
extern "C" void kernel_launch(void* const* d_in, const int* in_sizes, int n_in,
                              void* d_out, int out_size, void* d_ws, size_t ws_size,
                              hipStream_t stream) {
  (void)in_sizes; (void)n_in; (void)out_size; (void)ws_size;
  const float* x    = (const float*)d_in[0];   // (8,256,128,128) f32
  const float* mask = (const float*)d_in[1];   // (8,1,128,128) f32
  float* out    = (float*)d_out;               // scalar f32
  float* counts = (float*)d_ws;                // B*NSEG floats (8 KB)

  mcl_init_kernel<<<(BB * NSEG + 255) / 256, 256, 0, stream>>>(counts, out);
  mcl_count_kernel<<<BB * 8, 256, 0, stream>>>(mask, counts);
  mcl_main_kernel<<<BB * CC, 256, 0, stream>>>(x, mask, counts, out);
}